// attention_decoder_77549929496628
// MI455X (gfx1250) — compile-verified
//
#include <hip/hip_runtime.h>

// CDNA5 (gfx1250) wave32 WMMA, fp32 path: V_WMMA_F32_16X16X4_F32.
typedef __attribute__((ext_vector_type(2))) float f32x2;
typedef __attribute__((ext_vector_type(8))) float f32x8;

#define LDA 130              // leading dim for 128-col stage buffers (bank-friendly, even)
#define LDB 66               // leading dim for <=64-col stage buffers
#define SLD 48               // leading dim for score buffer (tile-padded)
#define XSZ 4160             // max(32*130, 48*66): full-tile padded rows
#define SSZ (48 * SLD)       // 48 rows padded
#define TSZ (16 * 66)

// per-attention-block GEMM dims (9 blocks, 3 projections each)
constexpr int DIN[9]  = {128, 128, 64, 128, 128, 64, 64, 64, 16};
constexpr int DOUT[9] = {128, 64, 64, 128, 64, 64, 64, 16, 1};
// flat-float offsets of each transposed W (dout x din) in the weight scratch
constexpr int WOFF[27] = {
    0,      16384,  32768,        // state1 q,k,v (128x128)
    49152,  57344,  65536,        // state2 (128x64)
    73728,  77824,  81920,        // state3 (64x64)
    86016,  102400, 118784,       // ts1 (128x128)
    135168, 143360, 151552,       // ts2 (128x64)
    159744, 163840, 167936,       // ts3 (64x64)
    172032, 176128, 180224,       // out1 (64x64)
    184320, 185344, 186368,       // out2 (64x16)
    187392, 187408, 187424};      // out3 (16x1)
constexpr int WTOTAL = 187440 + 256;  // floats (+256 slack for B-tile col overrun)

// fallback weight scratch if d_ws is too small (selected by flag at runtime)
__device__ float g_wT[WTOTAL];

struct DecoderParams {
  const float* W[27];
  const float* b[27];
  const float* g[27];
  const float* bt[27];
  const float* linW;
  const float* linb;
};

__device__ __forceinline__ f32x8 wmma_f32(f32x2 a, f32x2 b, f32x8 c) {
  // 8-arg pattern: (neg_a, A, neg_b, B, c_mod, C, reuse_a, reuse_b)
  return __builtin_amdgcn_wmma_f32_16x16x4_f32(false, a, false, b, (short)0, c,
                                               false, false);
}

// Inline erf (Abramowitz-Stegun 7.1.26, |err| <= 1.5e-7); v_rcp keeps it pure cheap ALU.
__device__ __forceinline__ float erf_inl(float x) {
  const float ax = fabsf(x);
  const float t = __builtin_amdgcn_rcpf(fmaf(0.3275911f, ax, 1.0f));
  float p = fmaf(t, 1.061405429f, -1.453152027f);
  p = fmaf(t, p, 1.421413741f);
  p = fmaf(t, p, -0.284496736f);
  p = fmaf(t, p, 0.254829592f);
  const float y = 1.0f - (t * p) * __expf(-ax * ax);
  return copysignf(y, x);
}
__device__ __forceinline__ float gelu_exact(float h) {
  return 0.5f * h * (1.0f + erf_inl(h * 0.70710678118654752f));
}

// H = layernorm(gelu_exact(X @ W + bias)) -> O
// X: rows x din in LDS; W transposed (dout x din) -> contiguous b64 pairs.
// Buffers are tile-padded: all GEMM loads/stores are unconditional; garbage stays
// in pad rows/cols and is never contracted over.
__device__ __forceinline__ void proj_gelu_ln(const float* X, float* O,
                                             int rows, int din, int dout,
                                             int ldx, int ldo,
                                             const float* __restrict__ W,
                                             const float* __restrict__ bias,
                                             const float* __restrict__ gam,
                                             const float* __restrict__ beta) {
  const int lane = threadIdx.x;
  const int nsub = lane & 15;
  const int ksub = (lane >> 4) << 1;   // A/B K-pair: lanes 0-15 -> {0,1}, 16-31 -> {2,3}
  const int rsub = (lane >> 4) << 3;   // C/D rows: lanes 16-31 hold rows 8..15
  __builtin_prefetch(W, 0, 1);
  for (int m0 = 0; m0 < rows; m0 += 16) {
    const int M = m0 + nsub;
    for (int n0 = 0; n0 < dout; n0 += 16) {
      const int N = n0 + nsub;
      f32x8 acc = {0.f, 0.f, 0.f, 0.f, 0.f, 0.f, 0.f, 0.f};
#pragma unroll 4
      for (int k0 = 0; k0 < din; k0 += 4) {
        const int kk = k0 + ksub;
        f32x2 a, bf;
        a.x = X[M * ldx + kk];
        a.y = X[M * ldx + kk + 1];
        bf.x = W[N * din + kk];      // W[n][k], contiguous pair
        bf.y = W[N * din + kk + 1];
        acc = wmma_f32(a, bf, acc);
      }
      const float bv = (N < dout) ? bias[N] : 0.f;  // only guard: real OOB otherwise
#pragma unroll
      for (int r = 0; r < 8; ++r) {
        const int row = m0 + r + rsub;
        O[row * ldo + N] = gelu_exact(acc[r] + bv);  // unconditional store (padded)
      }
    }
  }
  __syncthreads();
  // row-wise LayerNorm (float2 fast path; dout==1 degenerates to scalar)
  const float invd = __builtin_amdgcn_rcpf((float)dout);  // dout is a power of 2 -> exact
  for (int row = lane; row < rows; row += 32) {
    float* rp = O + row * ldo;
    if ((dout & 1) == 0) {
      const int h = dout >> 1;
      const float2* rp2 = (const float2*)rp;
      float m = 0.f;
      for (int c = 0; c < h; ++c) { const float2 t = rp2[c]; m += t.x + t.y; }
      m *= invd;
      float v = 0.f;
      for (int c = 0; c < h; ++c) {
        const float2 t = rp2[c];
        const float dx = t.x - m, dy = t.y - m;
        v += dx * dx + dy * dy;
      }
      v *= invd;
      const float inv = rsqrtf(v + 1e-5f);
      const float2* g2 = (const float2*)gam;
      const float2* b2 = (const float2*)beta;
      float2* w2 = (float2*)rp;
      for (int c = 0; c < h; ++c) {
        const float2 t = rp2[c];
        const float2 gg = g2[c];
        const float2 bb = b2[c];
        float2 o;
        o.x = (t.x - m) * inv * gg.x + bb.x;
        o.y = (t.y - m) * inv * gg.y + bb.y;
        w2[c] = o;
      }
    } else {
      float m = 0.f;
      for (int c = 0; c < dout; ++c) m += rp[c];
      m *= invd;
      float v = 0.f;
      for (int c = 0; c < dout; ++c) { const float d = rp[c] - m; v += d * d; }
      v *= invd;
      const float inv = rsqrtf(v + 1e-5f);
      for (int c = 0; c < dout; ++c) rp[c] = (rp[c] - m) * inv * gam[c] + beta[c];
    }
  }
  __syncthreads();
}

// S = Q @ K^T   (rows x rows, feature dim dfeat)
__device__ __forceinline__ void qk_scores(const float* Q, const float* K, float* S,
                                          int rows, int dfeat, int ldq) {
  const int lane = threadIdx.x;
  const int nsub = lane & 15;
  const int ksub = (lane >> 4) << 1;
  const int rsub = (lane >> 4) << 3;
  for (int m0 = 0; m0 < rows; m0 += 16) {
    const int M = m0 + nsub;
    for (int n0 = 0; n0 < rows; n0 += 16) {
      const int N = n0 + nsub;
      f32x8 acc = {0.f, 0.f, 0.f, 0.f, 0.f, 0.f, 0.f, 0.f};
#pragma unroll 4
      for (int k0 = 0; k0 < dfeat; k0 += 4) {
        const int kk = k0 + ksub;
        f32x2 a, bf;
        a.x = Q[M * ldq + kk];
        a.y = Q[M * ldq + kk + 1];
        bf.x = K[N * ldq + kk];  // B[k][n] = K[n][k], contiguous pair
        bf.y = K[N * ldq + kk + 1];
        acc = wmma_f32(a, bf, acc);
      }
#pragma unroll
      for (int r = 0; r < 8; ++r) {
        const int row = m0 + r + rsub;
        S[row * SLD + N] = acc[r];  // unconditional (padded)
      }
    }
  }
  __syncthreads();
}

// rows is even (16/20/36) -> float2 row passes
__device__ __forceinline__ void softmax_rows(float* S, int rows, float scale) {
  const int lane = threadIdx.x;
  const int h = rows >> 1;
  for (int row = lane; row < rows; row += 32) {
    float2* rp2 = (float2*)(S + row * SLD);
    float mx = -1e30f;
    for (int c = 0; c < h; ++c) {
      const float2 t = rp2[c];
      mx = fmaxf(mx, fmaxf(t.x, t.y));
    }
    mx *= scale;
    float sum = 0.f;
    for (int c = 0; c < h; ++c) {
      float2 t = rp2[c];
      t.x = __expf(t.x * scale - mx);
      t.y = __expf(t.y * scale - mx);
      rp2[c] = t;
      sum += t.x + t.y;
    }
    const float inv = __builtin_amdgcn_rcpf(sum);  // sum >= 1 (max term = 1)
    for (int c = 0; c < h; ++c) {
      float2 t = rp2[c];
      t.x *= inv;
      t.y *= inv;
      rp2[c] = t;
    }
  }
  __syncthreads();
}

// O = S @ V   (rows x dout, contraction over rows; 16/20/36 are all multiples of 4)
__device__ __forceinline__ void attn_out(const float* S, const float* V, float* O,
                                         int rows, int dout, int ldv, int ldo) {
  const int lane = threadIdx.x;
  const int nsub = lane & 15;
  const int ksub = (lane >> 4) << 1;
  const int rsub = (lane >> 4) << 3;
  for (int m0 = 0; m0 < rows; m0 += 16) {
    const int M = m0 + nsub;
    for (int n0 = 0; n0 < dout; n0 += 16) {
      const int N = n0 + nsub;
      f32x8 acc = {0.f, 0.f, 0.f, 0.f, 0.f, 0.f, 0.f, 0.f};
#pragma unroll 4
      for (int k0 = 0; k0 < rows; k0 += 4) {
        const int kk = k0 + ksub;
        f32x2 a, bf;
        a.x = S[M * SLD + kk];
        a.y = S[M * SLD + kk + 1];
        bf.x = V[kk * ldv + N];
        bf.y = V[(kk + 1) * ldv + N];
        acc = wmma_f32(a, bf, acc);
      }
#pragma unroll
      for (int r = 0; r < 8; ++r) {
        const int row = m0 + r + rsub;
        O[row * ldo + N] = acc[r];  // unconditional (padded)
      }
    }
  }
  __syncthreads();
}

__device__ __forceinline__ void attn_block(float* X, float* Q, float* K, float* S,
                                           int rows, int din, int dout,
                                           int ldx, int ldo,
                                           const DecoderParams& P,
                                           const float* __restrict__ wsT, int blk) {
  const int pq = blk * 3 + 0, pk = blk * 3 + 1, pv = blk * 3 + 2;
  proj_gelu_ln(X, Q, rows, din, dout, ldx, ldo, wsT + WOFF[pq], P.b[pq], P.g[pq], P.bt[pq]);
  proj_gelu_ln(X, K, rows, din, dout, ldx, ldo, wsT + WOFF[pk], P.b[pk], P.g[pk], P.bt[pk]);
  qk_scores(Q, K, S, rows, dout, ldo);
  softmax_rows(S, rows, rsqrtf((float)dout));
  proj_gelu_ln(X, K, rows, din, dout, ldx, ldo, wsT + WOFF[pv], P.b[pv], P.g[pv], P.bt[pv]);  // V -> K
  attn_out(S, K, X, rows, dout, ldo, ldo);  // result becomes next X
}

// One-shot weight transpose: W (din x dout) -> wT (dout x din), one block per projection.
__global__ void transpose_weights_kernel(DecoderParams P, float* __restrict__ ws,
                                         int useWs) {
  float* base = useWs ? ws : g_wT;
  const int p = blockIdx.x;
  const int blk = p / 3;
  const int din = DIN[blk], dout = DOUT[blk];
  const float* __restrict__ W = P.W[p];
  float* __restrict__ WTp = base + WOFF[p];
  const int n = din * dout;
  for (int i = threadIdx.x; i < n; i += blockDim.x) {
    const int k = i / dout, c = i - k * dout;
    WTp[c * din + k] = W[i];
  }
}

__global__ __launch_bounds__(32, 4) void attention_decoder_kernel(
    const float* __restrict__ cur, const float* __restrict__ env,
    const float* __restrict__ tsin, DecoderParams P,
    const float* __restrict__ ws, int useWs, float* __restrict__ out, int B) {
  __shared__ float X[XSZ];
  __shared__ float Q[XSZ];
  __shared__ float K[XSZ];
  __shared__ float S[SSZ];
  __shared__ float TR[TSZ];  // stash for the 's' branch result (16 x 64)

  const float* wsT = useWs ? ws : (const float*)g_wT;
  const int b = blockIdx.x;
  const int lane = threadIdx.x;

  // ---- state branch: X = concat(current[b] (15x128), environment[b] (1x128))
  {
    const float2* cb = (const float2*)(cur + (size_t)b * 15 * 128);
    for (int i = lane; i < 15 * 64; i += 32) {
      const int r = i >> 6, c = (i & 63) << 1;
      *(float2*)&X[r * LDA + c] = cb[i];
    }
    const float2* eb = (const float2*)(env + (size_t)b * 128);
    for (int c = lane; c < 64; c += 32) *(float2*)&X[15 * LDA + (c << 1)] = eb[c];
  }
  __syncthreads();
  attn_block(X, Q, K, S, 16, 128, 128, LDA, LDA, P, wsT, 0);  // state1
  attn_block(X, Q, K, S, 16, 128, 64, LDA, LDB, P, wsT, 1);   // state2
  attn_block(X, Q, K, S, 16, 64, 64, LDB, LDB, P, wsT, 2);    // state3
  for (int i = lane; i < 16 * 32; i += 32) {
    const int r = i >> 5, c = (i & 31) << 1;
    *(float2*)&TR[r * 66 + c] = *(const float2*)&X[r * LDB + c];
  }
  __syncthreads();

  // ---- ts branch: X = ts[b] (20x128)
  {
    const float2* tb = (const float2*)(tsin + (size_t)b * 20 * 128);
    for (int i = lane; i < 20 * 64; i += 32) {
      const int r = i >> 6, c = (i & 63) << 1;
      *(float2*)&X[r * LDA + c] = tb[i];
    }
  }
  __syncthreads();
  attn_block(X, Q, K, S, 20, 128, 128, LDA, LDA, P, wsT, 3);  // ts1
  attn_block(X, Q, K, S, 20, 128, 64, LDA, LDB, P, wsT, 4);   // ts2
  attn_block(X, Q, K, S, 20, 64, 64, LDB, LDB, P, wsT, 5);    // ts3

  // ---- assemble o = concat(s (16x64), t (20x64)) -> 36 x 64 in X (ld LDB)
  for (int i = lane; i < 20 * 32; i += 32) {  // park t result in Q
    const int r = i >> 5, c = (i & 31) << 1;
    *(float2*)&Q[r * LDB + c] = *(const float2*)&X[r * LDB + c];
  }
  __syncthreads();
  for (int i = lane; i < 16 * 32; i += 32) {
    const int r = i >> 5, c = (i & 31) << 1;
    *(float2*)&X[r * LDB + c] = *(const float2*)&TR[r * 66 + c];
  }
  for (int i = lane; i < 20 * 32; i += 32) {
    const int r = i >> 5, c = (i & 31) << 1;
    *(float2*)&X[(16 + r) * LDB + c] = *(const float2*)&Q[r * LDB + c];
  }
  __syncthreads();
  attn_block(X, Q, K, S, 36, 64, 64, LDB, LDB, P, wsT, 6);  // out1
  attn_block(X, Q, K, S, 36, 64, 16, LDB, LDB, P, wsT, 7);  // out2
  attn_block(X, Q, K, S, 36, 16, 1, LDB, LDB, P, wsT, 8);   // out3

  // ---- final linear: o.reshape(36) @ lin_W (36x2) + lin_b ; mu, softplus(sigma)
  if (lane == 0) {
    float o0 = P.linb[0], o1 = P.linb[1];
    for (int i = 0; i < 36; ++i) {
      const float xi = X[i * LDB];  // dout==1 -> column 0
      o0 += xi * P.linW[i * 2 + 0];
      o1 += xi * P.linW[i * 2 + 1];
    }
    out[b] = o0;                                              // mu
    const float sp = (o1 > 20.f) ? o1 : log1pf(__expf(o1));   // softplus
    out[B + b] = sp + 1e-6f;                                  // sigma
  }
}

extern "C" void kernel_launch(void* const* d_in, const int* in_sizes, int n_in,
                              void* d_out, int out_size, void* d_ws, size_t ws_size,
                              hipStream_t stream) {
  (void)n_in; (void)out_size;
  const float* cur = (const float*)d_in[0];
  const float* env = (const float*)d_in[1];
  const float* tsb = (const float*)d_in[2];

  // params flattened in recursive insertion order:
  // blocks [state1,state2,state3,ts1,ts2,ts3,out1,out2,out3] x projs [q,k,v] x [W,b,g,beta]
  DecoderParams P;
  int idx = 3;
  for (int p = 0; p < 27; ++p) {
    P.W[p] = (const float*)d_in[idx++];
    P.b[p] = (const float*)d_in[idx++];
    P.g[p] = (const float*)d_in[idx++];
    P.bt[p] = (const float*)d_in[idx++];
  }
  P.linW = (const float*)d_in[idx++];
  P.linb = (const float*)d_in[idx++];

  const int B = in_sizes[0] / (15 * 128);
  const int useWs = (ws_size >= (size_t)WTOTAL * sizeof(float)) ? 1 : 0;
  float* ws = (float*)d_ws;

  transpose_weights_kernel<<<dim3(27), dim3(256), 0, stream>>>(P, ws, useWs);
  attention_decoder_kernel<<<dim3(B), dim3(32), 0, stream>>>(
      cur, env, tsb, P, ws, useWs, (float*)d_out, B);
}